// Chain_42159398978141
// MI455X (gfx1250) — compile-verified
//
#include <hip/hip_runtime.h>
#include <hip/hip_bf16.h>

// ---------------------------------------------------------------------------
// Max-product BP on a binary chain CRF.  B=4096 rows, L=4096 chain length.
// Memory-bound (~0.5 GB traffic vs ~0.7 GFLOP) -> optimize data movement:
//   * wave32 per row, 2x2 max-product matrix-composition scan across lanes
//   * async global->LDS streaming of the observation row (ASYNCcnt), double
//     buffered, via __builtin_amdgcn_global_load_async_to_lds_b128.
// ---------------------------------------------------------------------------

#define BP_L       4096
#define BP_B       4096
#define BP_CHUNK   1024          // elements staged per chunk (4 KB)
#define BP_SEG     32            // elements per lane
#define BP_NCHUNK  (BP_L / BP_CHUNK)
#define BP_WPB     8             // waves (rows) per 256-thread block

#if defined(__has_builtin)
#if __has_builtin(__builtin_amdgcn_global_load_async_to_lds_b128) && \
    __has_builtin(__builtin_amdgcn_s_wait_asynccnt)
#define USE_ASYNC_LDS 1
#else
#define USE_ASYNC_LDS 0
#endif
#else
#define USE_ASYNC_LDS 0
#endif

typedef __attribute__((ext_vector_type(4))) int v4i;

struct M2 { float a00, a01, a10, a11; };   // 2x2 max-product matrix

__device__ __forceinline__ M2 mcompose(M2 b, M2 a) {
  // (b o a)[k][l] = max_m b[k][m] * a[m][l]   (all entries >= 0)
  M2 r;
  r.a00 = fmaxf(b.a00 * a.a00, b.a01 * a.a10);
  r.a01 = fmaxf(b.a00 * a.a01, b.a01 * a.a11);
  r.a10 = fmaxf(b.a10 * a.a00, b.a11 * a.a10);
  r.a11 = fmaxf(b.a10 * a.a01, b.a11 * a.a11);
  return r;
}

__device__ __forceinline__ M2 mshfl_up(M2 s, int d) {
  M2 r;
  r.a00 = __shfl_up(s.a00, d, 32);
  r.a01 = __shfl_up(s.a01, d, 32);
  r.a10 = __shfl_up(s.a10, d, 32);
  r.a11 = __shfl_up(s.a11, d, 32);
  return r;
}

#if USE_ASYNC_LDS
__device__ __forceinline__ void async_chunk_copy(const int* gsrc, int* ldst,
                                                 int lane) {
  // Builtin signature (from clang diagnostic):
  //   (v4i __device__* /*AS1*/, v4i __shared__* /*AS3*/, int offset, int cpol)
  int* gnc = const_cast<int*>(gsrc);
  __attribute__((address_space(1))) v4i* g =
      (__attribute__((address_space(1))) v4i*)gnc;
  __attribute__((address_space(3))) v4i* l =
      (__attribute__((address_space(3))) v4i*)ldst;
#pragma unroll
  for (int jj = 0; jj < 8; ++jj) {            // 8 x (32 lanes x 16B) = 4 KB
    const int idx = jj * 32 + lane;           // v4i-granular (16 B) index
    __builtin_amdgcn_global_load_async_to_lds_b128(g + idx, l + idx, 0, 0);
  }
}
#endif

template <int REVERSE>
__global__ __launch_bounds__(256) void bp_chain_scan(
    const int* __restrict__ obs, const float* __restrict__ jp,
    const float* __restrict__ bp, float* __restrict__ out) {
  const int lane = threadIdx.x & 31;
  const int wib  = threadIdx.x >> 5;
  const int row  = blockIdx.x * BP_WPB + wib;

#if USE_ASYNC_LDS
  __shared__ int lbuf[BP_WPB][2][BP_CHUNK];   // 64 KB / block, per-wave region
#endif

  // Potentials (uniform per dispatch; scalarized by the compiler).
  const float j = jp[0];
  const float p = __expf(0.25f * j);          // psi diag
  const float q = __expf(-0.25f * j);         // psi off-diag
  const float f00 = __expf(-0.5f * bp[0]), f10 = __expf(0.5f * bp[0]);
  const float f01 = __expf(-0.5f * bp[1]), f11 = __expf(0.5f * bp[1]);
  // Per-observation step matrices M[k][l] = psi[k][l] * phi[l]
  const M2 T0 = {p * f00, q * f10, q * f00, p * f10};
  const M2 T1 = {p * f01, q * f11, q * f01, p * f11};

  const int* grow = obs + (size_t)row * BP_L;
  float* orow0 = out + (size_t)row * (2 * BP_L);
  float* orow1 = orow0 + BP_L;

  float c0 = 1.0f, c1 = 1.0f;                 // chunk-entry message

#if USE_ASYNC_LDS
  {
    const int w0 = REVERSE ? (BP_L - BP_CHUNK) : 0;
    async_chunk_copy(grow + w0, &lbuf[wib][0][0], lane);
  }
#endif

  for (int c = 0; c < BP_NCHUNK; ++c) {
    const int cb = c * BP_CHUNK;
#if USE_ASYNC_LDS
    if (c + 1 < BP_NCHUNK) {
      const int w0n = REVERSE ? (BP_L - (cb + 2 * BP_CHUNK)) : (cb + BP_CHUNK);
      async_chunk_copy(grow + w0n, &lbuf[wib][(c + 1) & 1][0], lane);
      __builtin_amdgcn_s_wait_asynccnt(8);    // chunk c's 8 copies are done
    } else {
      __builtin_amdgcn_s_wait_asynccnt(0);
    }
    const int* lsrc = &lbuf[wib][c & 1][0];
#endif
    auto geto = [&](int ql) -> int {          // ql in [0, BP_CHUNK)
#if USE_ASYNC_LDS
      return lsrc[REVERSE ? (BP_CHUNK - 1 - ql) : ql];
#else
      return grow[REVERSE ? (BP_L - 1 - cb - ql) : (cb + ql)];
#endif
    };

    // ---- Phase 1: per-lane segment composite --------------------------
    M2 A = {1.f, 0.f, 0.f, 1.f};
#pragma unroll
    for (int k = 0; k < BP_SEG; ++k) {
      const int o = geto(lane * BP_SEG + k);
      M2 M;
      M.a00 = o ? T1.a00 : T0.a00;
      M.a01 = o ? T1.a01 : T0.a01;
      M.a10 = o ? T1.a10 : T0.a10;
      M.a11 = o ? T1.a11 : T0.a11;
      A = mcompose(M, A);
    }

    // ---- Phase 2: wave-level inclusive scan of composites -------------
    M2 S = A;
#pragma unroll
    for (int d = 1; d < 32; d <<= 1) {
      M2 T = mshfl_up(S, d);
      M2 C = mcompose(S, T);
      if (lane >= d) S = C;
    }
    M2 P = mshfl_up(S, 1);                    // exclusive prefix
    if (lane == 0) { P.a00 = 1.f; P.a01 = 0.f; P.a10 = 0.f; P.a11 = 1.f; }
    float m0 = fmaxf(P.a00 * c0, P.a01 * c1); // lane's entry message
    float m1 = fmaxf(P.a10 * c0, P.a11 * c1);
    const float s00 = __shfl(S.a00, 31, 32), s01 = __shfl(S.a01, 31, 32);
    const float s10 = __shfl(S.a10, 31, 32), s11 = __shfl(S.a11, 31, 32);
    const float nc0 = fmaxf(s00 * c0, s01 * c1);   // next chunk carry
    const float nc1 = fmaxf(s10 * c0, s11 * c1);

    // ---- Phase 3: replay segment, emit/fold output --------------------
#pragma unroll
    for (int g = 0; g < BP_SEG / 4; ++g) {
      float x0[4], x1[4], bm0[4], bm1[4];
#pragma unroll
      for (int e = 0; e < 4; ++e) {
        const int k = g * 4 + e;
        const int o = geto(lane * BP_SEG + k);
        const float f0 = o ? f01 : f00;
        const float f1 = o ? f11 : f10;
        bm0[e] = m0; bm1[e] = m1;             // message BEFORE consuming phi
        const float xx0 = f0 * m0;            // phi * msg  (== phi*fwd)
        const float xx1 = f1 * m1;
        x0[e] = xx0; x1[e] = xx1;
        m0 = fmaxf(p * xx0, q * xx1);         // step
        m1 = fmaxf(q * xx0, p * xx1);
      }
      if (!REVERSE) {
        const int t0 = cb + lane * BP_SEG + g * 4;
        *(float4*)(orow0 + t0) = make_float4(x0[0], x0[1], x0[2], x0[3]);
        *(float4*)(orow1 + t0) = make_float4(x1[0], x1[1], x1[2], x1[3]);
      } else {
        // logical q -> t = L-1-q; group occupies [tb, tb+3], e -> k = g*4+3-e
        const int tb = BP_L - 4 - (cb + lane * BP_SEG + g * 4);
        float4 a0 = *(float4*)(orow0 + tb);
        float4 a1 = *(float4*)(orow1 + tb);
        a0.x *= bm0[3]; a0.y *= bm0[2]; a0.z *= bm0[1]; a0.w *= bm0[0];
        a1.x *= bm1[3]; a1.y *= bm1[2]; a1.z *= bm1[1]; a1.w *= bm1[0];
        *(float4*)(orow0 + tb) = a0;
        *(float4*)(orow1 + tb) = a1;
      }
    }
    c0 = nc0; c1 = nc1;
  }
}

extern "C" void kernel_launch(void* const* d_in, const int* in_sizes, int n_in,
                              void* d_out, int out_size, void* d_ws,
                              size_t ws_size, hipStream_t stream) {
  (void)in_sizes; (void)n_in; (void)out_size; (void)d_ws; (void)ws_size;
  const float* jp  = (const float*)d_in[0];   // scalar j
  const float* bp  = (const float*)d_in[1];   // b[2]
  const int*   obs = (const int*)d_in[2];     // [B, L] int32
  float*       out = (float*)d_out;           // [B, 2, L] float32

  dim3 block(256);
  dim3 grid(BP_B / BP_WPB);
  // Pass 1: out[b,s,t] = phi[t,s] * fwd[t,s]
  bp_chain_scan<0><<<grid, block, 0, stream>>>(obs, jp, bp, out);
  // Pass 2: out[b,s,t] *= bwd[t,s]
  bp_chain_scan<1><<<grid, block, 0, stream>>>(obs, jp, bp, out);
}